// FurtherAttention_87067577024930
// MI455X (gfx1250) — compile-verified
//
#include <hip/hip_runtime.h>
#include <math.h>

// Problem constants (from reference): S=1024, B=64, D=1024, O=512
constexpr int S_ = 1024;
constexpr int B_ = 64;
constexpr int D_ = 1024;
constexpr int O_ = 512;

typedef __attribute__((ext_vector_type(2))) float v2f;
typedef __attribute__((ext_vector_type(8))) float v8f;

// ---------------------------------------------------------------------------
// Kernel 1a: lengths[b] = sum_s x1[s,b,0]  (mask is constant along d)
// ---------------------------------------------------------------------------
__global__ void klen(const float* __restrict__ x1, float* __restrict__ inv_len) {
    int b = threadIdx.x;                 // 64 threads
    float s = 0.0f;
    for (int i = 0; i < S_; ++i)
        s += x1[(size_t)i * (B_ * D_) + (size_t)b * D_];
    inv_len[b] = 1.0f / s;
}

// ---------------------------------------------------------------------------
// Kernel 1b: x_avg[b,d] = (sum_s x0[s,b,d]) * inv_len[b]   (streaming pass #1)
// ---------------------------------------------------------------------------
__global__ void kavg(const float* __restrict__ x0, const float* __restrict__ inv_len,
                     float* __restrict__ x_avg) {
    int idx = blockIdx.x * blockDim.x + threadIdx.x;    // idx = b*D + d
    float sum = 0.0f;
    for (int s = 0; s < S_; ++s)
        sum += x0[(size_t)s * (B_ * D_) + idx];
    x_avg[idx] = sum * inv_len[idx >> 10];              // idx / D_
}

// ---------------------------------------------------------------------------
// WMMA f32 GEMM:  Out[m,n] = relu?( sum_k A[m,k]*W[n,k] + bias[n] )
// One wave computes one 16x16 tile of Out using V_WMMA_F32_16X16X4_F32,
// marching K in steps of 4.
//
// VGPR layouts per CDNA5 ISA 7.12.2 (wave32):
//   A 16x4:  lanes 0-15 -> M=lane, {K=k0,k0+1}; lanes 16-31 -> M=lane-16, {K=k0+2,k0+3}
//   B 4x16:  lanes 0-15 -> N=lane, {K=k0,k0+1}; lanes 16-31 -> N=lane-16, {K=k0+2,k0+3}
//   C/D:     VGPR v: lanes 0-15 -> M=m0+v, N=lane; lanes 16-31 -> M=m0+v+8, N=lane-16
// Both A and B fetches are 2 consecutive f32 per lane -> 8B global loads.
// ---------------------------------------------------------------------------
__global__ void wmma_gemm_bias(const float* __restrict__ A, const float* __restrict__ W,
                               const float* __restrict__ bias, float* __restrict__ Out,
                               int M, int N, int K, int relu) {
    const int lane   = threadIdx.x & 31;
    const int wave   = threadIdx.x >> 5;
    const int wavesPerBlock = blockDim.x >> 5;
    const int tile   = blockIdx.x * wavesPerBlock + wave;
    const int tilesN = N >> 4;
    const int tm = tile / tilesN;
    const int tn = tile % tilesN;
    if (tm >= (M >> 4)) return;         // wave-uniform: whole wave exits together
    const int m0 = tm << 4;
    const int n0 = tn << 4;

    const int half = lane >> 4;         // 0: K=k0,k0+1  1: K=k0+2,k0+3
    const int l15  = lane & 15;
    const float* Arow = A + (size_t)(m0 + l15) * K + (half << 1);
    const float* Wrow = W + (size_t)(n0 + l15) * K + (half << 1);

    v8f acc = {};
    #pragma unroll 4
    for (int k0 = 0; k0 < K; k0 += 4) {
        v2f av = *(const v2f*)(Arow + k0);
        v2f bv = *(const v2f*)(Wrow + k0);
        // 8 args: (neg_a, A, neg_b, B, c_mod, C, reuse_a, reuse_b)
        acc = __builtin_amdgcn_wmma_f32_16x16x4_f32(
            false, av, false, bv, (short)0, acc, false, false);
    }

    const int col   = n0 + l15;
    const float bc  = bias[col];
    const int rbase = m0 + (half << 3);
    #pragma unroll
    for (int v = 0; v < 8; ++v) {
        float r = acc[v] + bc;
        if (relu) r = fmaxf(r, 0.0f);
        Out[(size_t)(rbase + v) * N + col] = r;
    }
}

// ---------------------------------------------------------------------------
// Kernel 3: e[s,b] = sum_d x0[s,b,d] * q[b,d]    (streaming pass #2)
// One wave per (s,b); lanes stride float4 across D, then wave reduction.
// ---------------------------------------------------------------------------
__global__ void kdot(const float* __restrict__ x0, const float* __restrict__ q,
                     float* __restrict__ e) {
    const int lane = threadIdx.x & 31;
    const int wid  = (blockIdx.x * blockDim.x + threadIdx.x) >> 5;  // s*B + b
    const int s = wid >> 6;        // / B_
    const int b = wid & (B_ - 1);
    const float* xr = x0 + (size_t)s * (B_ * D_) + (size_t)b * D_;
    const float* qr = q + (size_t)b * D_;
    float sum = 0.0f;
    #pragma unroll
    for (int i = 0; i < D_; i += 128) {
        float4 xv = *(const float4*)(xr + i + lane * 4);
        float4 qv = *(const float4*)(qr + i + lane * 4);
        sum += xv.x * qv.x + xv.y * qv.y + xv.z * qv.z + xv.w * qv.w;
    }
    #pragma unroll
    for (int off = 16; off; off >>= 1)
        sum += __shfl_xor(sum, off, 32);
    if (lane == 0) e[s * B_ + b] = sum;
}

// ---------------------------------------------------------------------------
// Kernel 4: masked softmax over s (per column b). a[s,b] written to output.
// ---------------------------------------------------------------------------
__global__ void ksoftmax(const float* __restrict__ e, float* __restrict__ a) {
    const int b   = blockIdx.x;     // 64 blocks
    const int tid = threadIdx.x;    // 256 threads, 4 s-values each
    __shared__ float red[256];

    float ev[4];
    float mx = -INFINITY;
    #pragma unroll
    for (int i = 0; i < 4; ++i) {
        float val = e[(tid + i * 256) * B_ + b];
        ev[i] = (val == 0.0f) ? -INFINITY : val;   // mask: e==0 -> -inf
        mx = fmaxf(mx, ev[i]);
    }
    red[tid] = mx; __syncthreads();
    for (int o = 128; o; o >>= 1) {
        if (tid < o) red[tid] = fmaxf(red[tid], red[tid + o]);
        __syncthreads();
    }
    mx = red[0]; __syncthreads();

    float sum = 0.0f;
    #pragma unroll
    for (int i = 0; i < 4; ++i) {
        ev[i] = (ev[i] == -INFINITY) ? 0.0f : __expf(ev[i] - mx);
        sum += ev[i];
    }
    red[tid] = sum; __syncthreads();
    for (int o = 128; o; o >>= 1) {
        if (tid < o) red[tid] += red[tid + o];
        __syncthreads();
    }
    const float inv = 1.0f / red[0];
    #pragma unroll
    for (int i = 0; i < 4; ++i)
        a[(tid + i * 256) * B_ + b] = ev[i] * inv;
}

// ---------------------------------------------------------------------------
// Kernel 5: xa[b,d] = sum_s a[s,b] * x0[s,b,d]   (streaming pass #3)
// Block = (b, 256-wide d-chunk); 'a' column staged through LDS.
// Key identity: c = xa @ Wv^T + bv  (softmax weights sum to 1), so the
// full (S,B,D) v-GEMM is never materialized.
// ---------------------------------------------------------------------------
__global__ void kwsum(const float* __restrict__ x0, const float* __restrict__ a,
                      float* __restrict__ xa) {
    const int chunks = D_ / 256;
    const int b  = blockIdx.x / chunks;
    const int dc = blockIdx.x % chunks;
    const int d  = dc * 256 + threadIdx.x;
    __shared__ float aS[256];

    float acc = 0.0f;
    const float* xp = x0 + (size_t)b * D_ + d;
    for (int s0 = 0; s0 < S_; s0 += 256) {
        aS[threadIdx.x] = a[(s0 + threadIdx.x) * B_ + b];
        __syncthreads();
        #pragma unroll 8
        for (int i = 0; i < 256; ++i)
            acc += aS[i] * xp[(size_t)(s0 + i) * (B_ * D_)];
        __syncthreads();
    }
    xa[b * D_ + d] = acc;
}

// ---------------------------------------------------------------------------
extern "C" void kernel_launch(void* const* d_in, const int* in_sizes, int n_in,
                              void* d_out, int out_size, void* d_ws, size_t ws_size,
                              hipStream_t stream) {
    const float* x0 = (const float*)d_in[0];
    const float* x1 = (const float*)d_in[1];
    const float* Wq = (const float*)d_in[2];
    const float* bq = (const float*)d_in[3];
    const float* Wv = (const float*)d_in[4];
    const float* bv = (const float*)d_in[5];
    const float* Wc = (const float*)d_in[6];
    const float* bc = (const float*)d_in[7];

    float* out_p = (float*)d_out;              // (B,O) = 32768 floats
    float* a_out = out_p + B_ * O_;            // (S,B,1) = 65536 floats

    float* ws      = (float*)d_ws;             // ~1.3 MB total scratch
    float* inv_len = ws;                       // 64
    float* x_avg   = inv_len + 64;             // 65536
    float* qbuf    = x_avg + B_ * D_;          // 65536
    float* ebuf    = qbuf + B_ * D_;           // 65536
    float* xabuf   = ebuf + S_ * B_;           // 65536
    float* cbuf    = xabuf + B_ * D_;          // 65536

    // 1) averages (pass #1 over x0)
    klen<<<1, 64, 0, stream>>>(x1, inv_len);
    kavg<<<(B_ * D_) / 256, 256, 0, stream>>>(x0, inv_len, x_avg);

    // 2) q = x_avg @ Wq^T + bq          (64x1024x1024, WMMA f32)
    wmma_gemm_bias<<<(4 * (D_ / 16)) / 4, 128, 0, stream>>>(
        x_avg, Wq, bq, qbuf, B_, D_, D_, 0);

    // 3) e = <x0, q> along d            (pass #2 over x0)
    kdot<<<(S_ * B_) / 8, 256, 0, stream>>>(x0, qbuf, ebuf);

    // 4) a = masked softmax_s(e)  -> directly into output slot
    ksoftmax<<<B_, 256, 0, stream>>>(ebuf, a_out);

    // 5) xa = sum_s a * x0              (pass #3 over x0)
    kwsum<<<B_ * (D_ / 256), 256, 0, stream>>>(x0, a_out, xabuf);

    // 6) c = xa @ Wv^T + bv             (64x1024x1024, WMMA f32)
    wmma_gemm_bias<<<(4 * (D_ / 16)) / 4, 128, 0, stream>>>(
        xabuf, Wv, bv, cbuf, B_, D_, D_, 0);

    // 7) out = relu(c @ Wc^T + bc)      (64x512x1024, WMMA f32)
    wmma_gemm_bias<<<(4 * (O_ / 16)) / 4, 128, 0, stream>>>(
        cbuf, Wc, bc, out_p, B_, O_, D_, 1);
}